// ExpHydroM100_22445499089624
// MI455X (gfx1250) — compile-verified
//
#include <hip/hip_runtime.h>
#include <hip/hip_bf16.h>

typedef __attribute__((ext_vector_type(2))) float v2f;
typedef __attribute__((ext_vector_type(8))) float v8f;

#define T_LEN 2048
#define B_LEN 64
#define HS 68   // padded row stride for 64-wide LDS matrices (68*4B -> bank-conflict-free frags)
#define OS 20   // padded row stride for the 16-wide output tile

#define LOG2E 1.4426950408889634f

// ---- branchless fast transcendentals (native V_EXP_F32 / V_RCP_F32) ---------
__device__ __forceinline__ float fast_exp(float x) {
    x = fminf(fmaxf(x, -87.0f), 87.0f);                  // keep finite, no inf paths
    return __builtin_amdgcn_exp2f(x * LOG2E);
}
__device__ __forceinline__ float fast_tanh(float x) {
    float xc = fminf(fmaxf(x, -9.0f), 9.0f);             // tanh(±9) == ±1 in f32
    float r = __builtin_amdgcn_exp2f(xc * (2.0f * LOG2E));  // e^{2x}
    return (r - 1.0f) * __builtin_amdgcn_rcpf(r + 1.0f);
}
__device__ __forceinline__ float fast_sinh(float x) {
    float xc = fminf(fmaxf(x, -87.0f), 87.0f);
    float r = __builtin_amdgcn_exp2f(xc * LOG2E);        // e^{x}
    return 0.5f * (r - __builtin_amdgcn_rcpf(r));
}
__device__ __forceinline__ float stepf(float x) { return 0.5f * (fast_tanh(5.0f * x) + 1.0f); }

// ---- WMMA fragment helpers (V_WMMA_F32_16X16X4_F32, fp32 in/out) ------------
// A (16x4): lane L -> row m=L&15, VGPR0/1 = K {koff, koff+1}, koff = (L>>4)*2
// B (4x16): lane L -> col n,       VGPR0/1 = K {koff, koff+1}   (stored transposed: Bt[n][k])
// C/D (16x16): VGPR v -> M = v + 8*(L>>4), N = L&15

__device__ __forceinline__ v8f wmma_tile_k4(const float* __restrict__ X /*16x4*/,
                                            const float* __restrict__ W1t /*64x4*/,
                                            int nbase, int lane) {
    const int m = lane & 15;
    const int koff = (lane >> 4) * 2;
    v2f a = *(const v2f*)(X + m * 4 + koff);
    v2f b = *(const v2f*)(W1t + (nbase + m) * 4 + koff);
    v8f acc = {};
    acc = __builtin_amdgcn_wmma_f32_16x16x4_f32(false, a, false, b, (short)0, acc, false, false);
    return acc;
}

__device__ __forceinline__ v8f wmma_tile_k64(const float* __restrict__ A /*16 x HS*/,
                                             const float* __restrict__ Bt /*64 x HS, Bt[n][k]*/,
                                             int nbase, int lane) {
    const int m = lane & 15;
    const int koff = (lane >> 4) * 2;
    const int nrow = (nbase + m) * HS;
    v8f acc = {};
#pragma unroll
    for (int kt = 0; kt < 16; ++kt) {
        const int kb = kt * 4 + koff;
        v2f a = *(const v2f*)(A + m * HS + kb);
        v2f b = *(const v2f*)(Bt + nrow + kb);
        acc = __builtin_amdgcn_wmma_f32_16x16x4_f32(false, a, false, b, (short)0, acc, false, false);
    }
    return acc;
}

__device__ __forceinline__ void store_tile(v8f acc, float* __restrict__ Hout, int stride,
                                           const float* __restrict__ bias, int nbase,
                                           int lane, bool act) {
    const int n = nbase + (lane & 15);
    const int mhi = (lane >> 4) * 8;
    const float bv = bias[n];
#pragma unroll
    for (int v = 0; v < 8; ++v) {
        float x = acc[v] + bv;
        if (act) x = fast_tanh(x);
        Hout[(mhi + v) * stride + n] = x;
    }
}

// Full 16-row MLP: X(16x4) -> tanh(.@W1+b1) -> tanh(.@W2+b2) -> tanh(.@W3+b3) -> .@W4+b4
// Weights resident in LDS; W2/W3/W4 stored transposed (Wt[n][k]), W4 zero-padded to N=16.
__device__ __forceinline__ void mlp_forward(const float* sX, const float* sW1t, const float* sb1,
                                            const float* sW2t, const float* sb2,
                                            const float* sW3t, const float* sb3,
                                            const float* sW4t, const float* sb4,
                                            float* sHa, float* sHb, float* sO,
                                            int lane, int wave) {
    v8f acc = wmma_tile_k4(sX, sW1t, wave * 16, lane);
    store_tile(acc, sHa, HS, sb1, wave * 16, lane, true);
    __syncthreads();
    acc = wmma_tile_k64(sHa, sW2t, wave * 16, lane);
    store_tile(acc, sHb, HS, sb2, wave * 16, lane, true);
    __syncthreads();
    acc = wmma_tile_k64(sHb, sW3t, wave * 16, lane);
    store_tile(acc, sHa, HS, sb3, wave * 16, lane, true);
    __syncthreads();
    if (wave == 0) {  // wave-uniform: EXEC stays all-ones inside the wave
        acc = wmma_tile_k64(sHa, sW4t, 0, lane);
        store_tile(acc, sO, OS, sb4, 0, lane, false);
    }
    __syncthreads();
}

#define LOAD_WEIGHTS_TO_LDS()                                                          \
    do {                                                                               \
        for (int i = tid; i < 16 * HS; i += 128) sW4t[i] = 0.0f;                       \
        if (tid < 16) sb4[tid] = 0.0f;                                                 \
        __syncthreads();                                                               \
        for (int i = tid; i < 4096; i += 128) {                                        \
            int k = i >> 6, n = i & 63;                                                \
            sW2t[n * HS + k] = W2[i];                                                  \
            sW3t[n * HS + k] = W3[i];                                                  \
        }                                                                              \
        for (int i = tid; i < 256; i += 128) {                                         \
            int k = i >> 6, n = i & 63;                                                \
            sW1t[n * 4 + k] = W1[i];                                                   \
        }                                                                              \
        for (int i = tid; i < 320; i += 128) {                                         \
            int k = i / 5, n = i % 5;                                                  \
            sW4t[n * HS + k] = W4[i];                                                  \
        }                                                                              \
        if (tid < 64) { sb1[tid] = b1[tid]; sb2[tid] = b2[tid]; sb3[tid] = b3[tid]; }  \
        if (tid < 5) sb4[tid] = b4[tid];                                               \
    } while (0)

// ---------------- Kernel 1: sequential RK4 scan, 16 independent batch rows/block ----
__global__ __launch_bounds__(128) void exphydro_scan_kernel(
    const float* __restrict__ s_snow, const float* __restrict__ s_water,
    const float* __restrict__ precp, const float* __restrict__ tmean,
    const float* __restrict__ lday, const float* __restrict__ times,
    const float* __restrict__ W1, const float* __restrict__ b1,
    const float* __restrict__ W2, const float* __restrict__ b2,
    const float* __restrict__ W3, const float* __restrict__ b3,
    const float* __restrict__ W4, const float* __restrict__ b4,
    float* __restrict__ traj /* B*T*2 */) {
    __shared__ __align__(16) float sW1t[64 * 4], sW2t[64 * HS], sW3t[64 * HS], sW4t[16 * HS];
    __shared__ float sb1[64], sb2[64], sb3[64], sb4[16];
    __shared__ __align__(16) float sX[16 * 4], sHa[16 * HS], sHb[16 * HS], sO[16 * OS];
    __shared__ float sy[16][2], sc0[16], sc1[16], sTemp[16], sLday[16];
    __shared__ float sK[4][16][2];
    __shared__ float sTD[2];

    const int tid = threadIdx.x;
    const int lane = tid & 31, wave = tid >> 5;
    const int rbase = blockIdx.x * 16;

    LOAD_WEIGHTS_TO_LDS();

    if (tid < 16) {
        const int R = rbase + tid;
        const float y0v = s_snow[(size_t)R * T_LEN];
        const float y1v = s_water[(size_t)R * T_LEN];
        sy[tid][0] = y0v;
        sy[tid][1] = y1v;
        traj[((size_t)R * T_LEN) * 2 + 0] = y0v;
        traj[((size_t)R * T_LEN) * 2 + 1] = y1v;
    }
    __syncthreads();

    for (int stp = 0; stp < T_LEN - 1; ++stp) {
        if (tid == 0) {
            const float t0 = times[stp];
            sTD[0] = t0;
            sTD[1] = times[stp + 1] - t0;
        }
        __syncthreads();
        const float t0 = sTD[0], dtv = sTD[1];

#pragma unroll 1
        for (int sub = 0; sub < 4; ++sub) {
            if (tid < 16) {
                const int r = tid, R = rbase + r;
                float tc, c0, c1;
                if (sub == 0) {
                    tc = t0; c0 = sy[r][0]; c1 = sy[r][1];
                } else if (sub == 1) {
                    tc = t0 + 0.5f * dtv;
                    c0 = sy[r][0] + 0.5f * dtv * sK[0][r][0];
                    c1 = sy[r][1] + 0.5f * dtv * sK[0][r][1];
                } else if (sub == 2) {
                    tc = t0 + 0.5f * dtv;
                    c0 = sy[r][0] + 0.5f * dtv * sK[1][r][0];
                    c1 = sy[r][1] + 0.5f * dtv * sK[1][r][1];
                } else {
                    tc = t0 + dtv;
                    c0 = sy[r][0] + dtv * sK[2][r][0];
                    c1 = sy[r][1] + dtv * sK[2][r][1];
                }
                int i0 = (int)floorf(tc);
                i0 = i0 < 0 ? 0 : (i0 > T_LEN - 2 ? T_LEN - 2 : i0);
                const float fr = tc - (float)i0;
                const size_t base = (size_t)R * T_LEN + (size_t)i0;
                const float pv = precp[base] * (1.0f - fr) + precp[base + 1] * fr;
                const float tv = tmean[base] * (1.0f - fr) + tmean[base + 1] * fr;
                const float lv = lday[base] * (1.0f - fr) + lday[base + 1] * fr;
                sc0[r] = c0; sc1[r] = c1; sTemp[r] = tv; sLday[r] = lv;
                sX[r * 4 + 0] = c0;
                sX[r * 4 + 1] = c1;
                sX[r * 4 + 2] = pv;
                sX[r * 4 + 3] = tv;
            }
            __syncthreads();

            mlp_forward(sX, sW1t, sb1, sW2t, sb2, sW3t, sb3, sW4t, sb4, sHa, sHb, sO, lane, wave);

            if (tid < 16) {
                const int r = tid;
                const float o0 = sO[r * OS + 0], o1 = sO[r * OS + 1], o2 = sO[r * OS + 2];
                const float o3 = sO[r * OS + 3], o4 = sO[r * OS + 4];
                const float tv = sTemp[r], lv = sLday[r], c0 = sc0[r], c1 = sc1[r];
                const float ps = fmaxf(fast_sinh(o0) * stepf(-tv), 0.0f);
                const float pr = fmaxf(fast_sinh(o1), 0.0f);
                const float mm = fmaxf(stepf(c0) * fast_sinh(o2), 0.0f);
                const float s1s = stepf(c1);
                const float et = s1s * fast_exp(o3) * lv;
                const float qq = s1s * fast_exp(o4);
                sK[sub][r][0] = ps - mm;
                sK[sub][r][1] = pr + mm - et - qq;
            }
            __syncthreads();
        }

        if (tid < 16) {
            const int r = tid, R = rbase + r;
            const float f = dtv * (1.0f / 6.0f);
            const float y0v = sy[r][0] + f * (sK[0][r][0] + 2.0f * sK[1][r][0] + 2.0f * sK[2][r][0] + sK[3][r][0]);
            const float y1v = sy[r][1] + f * (sK[0][r][1] + 2.0f * sK[1][r][1] + 2.0f * sK[2][r][1] + sK[3][r][1]);
            sy[r][0] = y0v;
            sy[r][1] = y1v;
            const size_t gi = ((size_t)R * T_LEN + (size_t)(stp + 1)) * 2;
            traj[gi + 0] = y0v;
            traj[gi + 1] = y1v;
        }
        __syncthreads();
    }
}

// ---------------- Kernel 2: parallel output MLP over all B*T rows -------------------
__global__ __launch_bounds__(128) void exphydro_out_kernel(
    const float* __restrict__ precp, const float* __restrict__ tmean,
    const float* __restrict__ W1, const float* __restrict__ b1,
    const float* __restrict__ W2, const float* __restrict__ b2,
    const float* __restrict__ W3, const float* __restrict__ b3,
    const float* __restrict__ W4, const float* __restrict__ b4,
    const float* __restrict__ traj, float* __restrict__ out) {
    __shared__ __align__(16) float sW1t[64 * 4], sW2t[64 * HS], sW3t[64 * HS], sW4t[16 * HS];
    __shared__ float sb1[64], sb2[64], sb3[64], sb4[16];
    __shared__ __align__(16) float sX[16 * 4], sHa[16 * HS], sHb[16 * HS], sO[16 * OS];

    const int tid = threadIdx.x;
    const int lane = tid & 31, wave = tid >> 5;
    const int g0 = blockIdx.x * 16;

    LOAD_WEIGHTS_TO_LDS();

    if (tid < 16) {
        const size_t g = (size_t)(g0 + tid);
        sX[tid * 4 + 0] = traj[g * 2 + 0];
        sX[tid * 4 + 1] = traj[g * 2 + 1];
        sX[tid * 4 + 2] = precp[g];
        sX[tid * 4 + 3] = tmean[g];
    }
    __syncthreads();

    mlp_forward(sX, sW1t, sb1, sW2t, sb2, sW3t, sb3, sW4t, sb4, sHa, sHb, sO, lane, wave);

    if (tid < 16) out[g0 + tid] = sO[tid * OS + 4];
}

extern "C" void kernel_launch(void* const* d_in, const int* in_sizes, int n_in,
                              void* d_out, int out_size, void* d_ws, size_t ws_size,
                              hipStream_t stream) {
    const float* s_snow  = (const float*)d_in[0];
    const float* s_water = (const float*)d_in[1];
    const float* precp   = (const float*)d_in[2];
    const float* tmean   = (const float*)d_in[3];
    const float* lday    = (const float*)d_in[4];
    const float* times   = (const float*)d_in[5];
    const float* W1 = (const float*)d_in[6];
    const float* b1 = (const float*)d_in[7];
    const float* W2 = (const float*)d_in[8];
    const float* b2 = (const float*)d_in[9];
    const float* W3 = (const float*)d_in[10];
    const float* b3 = (const float*)d_in[11];
    const float* W4 = (const float*)d_in[12];
    const float* b4 = (const float*)d_in[13];
    float* traj = (float*)d_ws;  // B*T*2 floats = 1 MiB
    float* out = (float*)d_out;

    exphydro_scan_kernel<<<B_LEN / 16, 128, 0, stream>>>(
        s_snow, s_water, precp, tmean, lday, times,
        W1, b1, W2, b2, W3, b3, W4, b4, traj);

    exphydro_out_kernel<<<(B_LEN * T_LEN) / 16, 128, 0, stream>>>(
        precp, tmean, W1, b1, W2, b2, W3, b3, W4, b4, traj, out);
}